// GCN_71073118814860
// MI455X (gfx1250) — compile-verified
//
#include <hip/hip_runtime.h>
#include <math.h>

// Problem dims (fixed by the reference): F_IN=256, HID=128, C=40.
#define F_IN 256
#define HID  128
#define NCLS 40
#define LDS_STRIDE1 260   // 256 + 4 pad: 16 distinct banks per fragment, 16B rows
#define LDS_STRIDE2 132   // 128 + 4 pad: same property for GEMM2 tiles

typedef __attribute__((ext_vector_type(2))) float v2f;
typedef __attribute__((ext_vector_type(8))) float v8f;

// ---------------------------------------------------------------------------
// Zero-fill scratch (deterministic per call; harness poisons ws once)
// ---------------------------------------------------------------------------
__global__ __launch_bounds__(256) void zero_f32(float* __restrict__ p, long long n) {
    long long i = (long long)blockIdx.x * blockDim.x + threadIdx.x;
    long long stride = (long long)gridDim.x * blockDim.x;
    for (; i < n; i += stride) p[i] = 0.0f;
}

// ---------------------------------------------------------------------------
// GEMM1: H[N,128] = X[N,256] @ W1[256,128] via V_WMMA_F32_16X16X4_F32.
// Block = 256 threads (8 wave32s) -> 32x128 output tile.
//  * 32x256 A tile staged once in LDS (33KB, padded stride 260 floats).
//  * Wave w owns column tile w (16 cols) and both 16-row tiles: per K=4
//    step: 1 B fragment (global, L2-resident W1), 2 A fragments
//    (ds_load_2addr_b64), 2 WMMAs.  A reuse 8x via LDS, B reuse 2x in-reg.
// Fragment maps (wave32):
//  A 16x4 f32: lane l -> M = l&15, K = (l>>4)*2 + {0,1}
//  B 4x16 f32: lane l -> N = l&15, K = (l>>4)*2 + {0,1}
//  C/D 16x16:  lane l, vgpr v -> M = v + 8*(l>>4), N = l&15
// ---------------------------------------------------------------------------
__global__ __launch_bounds__(256) void gemm1_wmma(const float* __restrict__ X,
                                                  const float* __restrict__ W,
                                                  float* __restrict__ H) {
    __shared__ float As[32 * LDS_STRIDE1];

    const int tid  = threadIdx.x;
    const int row0 = blockIdx.x * 32;

    // Cooperative load: 32 rows x 256 floats = 2048 float4s, 8 per thread.
    const float4* __restrict__ Xv = (const float4*)(X + (size_t)row0 * F_IN);
#pragma unroll
    for (int i = 0; i < 8; ++i) {
        const int c  = tid + 256 * i;      // 0..2047
        const int r  = c >> 6;             // row in tile (64 float4 per row)
        const int kq = c & 63;             // float4 index within row
        const float4 v = Xv[(size_t)r * (F_IN / 4) + kq];
        *(float4*)&As[r * LDS_STRIDE1 + kq * 4] = v;
    }
    __syncthreads();

    const int lane = tid & 31;
    const int wave = tid >> 5;            // 0..7 -> column tile
    const int col0 = wave * 16;
    const int m    = lane & 15;
    const int kb   = (lane >> 4) << 1;    // 0 or 2
    const int n    = lane & 15;

    const float* __restrict__ a0p  = &As[m * LDS_STRIDE1 + kb];
    const float* __restrict__ a1p  = &As[(m + 16) * LDS_STRIDE1 + kb];
    const float* __restrict__ bcol = W + (size_t)kb * HID + (col0 + n);

    v8f c0 = {}, c1 = {};
#pragma unroll 4
    for (int k0 = 0; k0 < F_IN; k0 += 4) {
        v2f a0, a1, b;
        a0.x = a0p[k0];
        a0.y = a0p[k0 + 1];
        a1.x = a1p[k0];
        a1.y = a1p[k0 + 1];
        b.x  = bcol[(size_t)k0 * HID];
        b.y  = bcol[(size_t)k0 * HID + HID];
        c0 = __builtin_amdgcn_wmma_f32_16x16x4_f32(false, a0, false, b,
                                                   (short)0, c0, false, false);
        c1 = __builtin_amdgcn_wmma_f32_16x16x4_f32(false, a1, false, b,
                                                   (short)0, c1, false, false);
    }

    const int mBase = (lane >> 4) << 3;   // 0 or 8
    float* __restrict__ hp0 = H + (size_t)(row0 + mBase) * HID + col0 + n;
    float* __restrict__ hp1 = H + (size_t)(row0 + 16 + mBase) * HID + col0 + n;
#pragma unroll
    for (int v = 0; v < 8; ++v) {
        hp0[(size_t)v * HID] = c0[v];
        hp1[(size_t)v * HID] = c1[v];
    }
}

// ---------------------------------------------------------------------------
// GEMM2 (fused bias+ReLU): O[N,40] = relu(AGG1 + b1) @ W2[128,40].
// Block = 192 threads (6 wave32s) -> 32x40 output tile.
//  * relu(agg1 + b1) applied ONCE per element while staging the 32x128 A
//    tile into LDS (17KB, padded stride 132) -- kills the separate
//    bias_relu pass (saves ~100MB of traffic).
//  * Wave w: row tile w/3, column tile w%3 (48 cols; col >= 40 masked
//    branchlessly in loads + in stores after the last WMMA, EXEC all-ones).
// ---------------------------------------------------------------------------
__global__ __launch_bounds__(192) void gemm2_wmma(const float* __restrict__ AGG1,
                                                  const float* __restrict__ B1,
                                                  const float* __restrict__ W,
                                                  float* __restrict__ O) {
    __shared__ float As[32 * LDS_STRIDE2];

    const int tid  = threadIdx.x;
    const int row0 = blockIdx.x * 32;

    // Stage 32 rows x 128 floats = 1024 float4s with fused bias + ReLU.
    const float4* __restrict__ Av  = (const float4*)(AGG1 + (size_t)row0 * HID);
    const float4* __restrict__ B1v = (const float4*)B1;
    for (int c = tid; c < 1024; c += 192) {
        const int r  = c >> 5;             // row (32 float4 per row)
        const int kq = c & 31;             // float4 index within row
        float4 v = Av[(size_t)r * (HID / 4) + kq];
        const float4 bb = B1v[kq];
        v.x += bb.x; v.y += bb.y; v.z += bb.z; v.w += bb.w;
        v.x = v.x > 0.0f ? v.x : 0.0f;
        v.y = v.y > 0.0f ? v.y : 0.0f;
        v.z = v.z > 0.0f ? v.z : 0.0f;
        v.w = v.w > 0.0f ? v.w : 0.0f;
        *(float4*)&As[r * LDS_STRIDE2 + kq * 4] = v;
    }
    __syncthreads();

    const int lane   = tid & 31;
    const int wave   = tid >> 5;          // 0..5
    const int rowSel = wave / 3;          // 0 or 1 -> 16-row tile
    const int colT   = wave % 3;          // 0..2   -> 16-col tile
    const int col0   = colT * 16;
    const int m      = lane & 15;
    const int kb     = (lane >> 4) << 1;
    const int n      = lane & 15;
    const int colv   = col0 + n;
    const bool vld   = (colv < NCLS);

    const float* __restrict__ ap = &As[(rowSel * 16 + m) * LDS_STRIDE2 + kb];

    v8f c = {};
#pragma unroll 4
    for (int k0 = 0; k0 < HID; k0 += 4) {
        v2f a, b;
        a.x = ap[k0];
        a.y = ap[k0 + 1];
        b.x = vld ? W[(size_t)(k0 + kb) * NCLS + colv] : 0.0f;
        b.y = vld ? W[(size_t)(k0 + kb + 1) * NCLS + colv] : 0.0f;
        c = __builtin_amdgcn_wmma_f32_16x16x4_f32(false, a, false, b,
                                                  (short)0, c, false, false);
    }

    const int mBase = (lane >> 4) << 3;
    if (vld) {
        float* __restrict__ op =
            O + (size_t)(row0 + rowSel * 16 + mBase) * NCLS + colv;
#pragma unroll
        for (int v = 0; v < 8; ++v)
            op[(size_t)v * NCLS] = c[v];
    }
}

// ---------------------------------------------------------------------------
// Scatter1: one wave per edge. Lane l handles channels [4l..4l+3] (HID=128).
// Gathers stay in the 192MB L2 (h table = 51MB); atomics are L2-resident.
// ---------------------------------------------------------------------------
__global__ __launch_bounds__(256) void scatter1(const float* __restrict__ H,
                                                const int* __restrict__ src,
                                                const int* __restrict__ dst,
                                                const float* __restrict__ ew,
                                                float* __restrict__ AGG,
                                                int nEdges) {
    const int warp = (int)((blockIdx.x * (unsigned)blockDim.x + threadIdx.x) >> 5);
    const int lane = threadIdx.x & 31;
    if (warp >= nEdges) return;
    const int   s = src[warp];
    const int   d = dst[warp];
    const float w = ew[warp];
    const float4 v = ((const float4*)(H + (size_t)s * HID))[lane];
    float* __restrict__ a = AGG + (size_t)d * HID + lane * 4;
    atomicAdd(a + 0, v.x * w);
    atomicAdd(a + 1, v.y * w);
    atomicAdd(a + 2, v.z * w);
    atomicAdd(a + 3, v.w * w);
}

// ---------------------------------------------------------------------------
// Scatter2: one thread per (edge, class) pair; C=40.
// ---------------------------------------------------------------------------
__global__ __launch_bounds__(256) void scatter2(const float* __restrict__ O,
                                                const int* __restrict__ src,
                                                const int* __restrict__ dst,
                                                const float* __restrict__ ew,
                                                float* __restrict__ AGG,
                                                long long total) {
    long long i = (long long)blockIdx.x * blockDim.x + threadIdx.x;
    if (i >= total) return;
    const int e = (int)(i / NCLS);
    const int c = (int)(i - (long long)e * NCLS);
    const int s = src[e];
    const int d = dst[e];
    atomicAdd(AGG + (size_t)d * NCLS + c, O[(size_t)s * NCLS + c] * ew[e]);
}

// ---------------------------------------------------------------------------
// Row-wise log_softmax over 40 classes (after adding b2). One thread per row.
// ---------------------------------------------------------------------------
__global__ __launch_bounds__(256) void logsoftmax(const float* __restrict__ A,
                                                  const float* __restrict__ b,
                                                  float* __restrict__ OUT,
                                                  int nRows) {
    const int row = blockIdx.x * blockDim.x + threadIdx.x;
    if (row >= nRows) return;
    const float* __restrict__ a = A + (size_t)row * NCLS;
    float vals[NCLS];
    float mx = -INFINITY;
#pragma unroll
    for (int c = 0; c < NCLS; ++c) {
        vals[c] = a[c] + b[c];
        mx = vals[c] > mx ? vals[c] : mx;
    }
    float sum = 0.0f;
#pragma unroll
    for (int c = 0; c < NCLS; ++c)
        sum += __expf(vals[c] - mx);
    const float lse = __logf(sum) + mx;
    float* __restrict__ o = OUT + (size_t)row * NCLS;
#pragma unroll
    for (int c = 0; c < NCLS; ++c)
        o[c] = vals[c] - lse;
}

// ---------------------------------------------------------------------------
// Launch: zero -> gemm1 -> scatter1 -> gemm2(fused relu) -> scatter2 -> lsm
// ---------------------------------------------------------------------------
extern "C" void kernel_launch(void* const* d_in, const int* in_sizes, int n_in,
                              void* d_out, int out_size, void* d_ws, size_t ws_size,
                              hipStream_t stream) {
    const float* x  = (const float*)d_in[0];
    const int*   ei = (const int*)d_in[1];   // [2,E] flat: src then dst
    const float* ew = (const float*)d_in[2];
    const float* W1 = (const float*)d_in[3];
    const float* b1 = (const float*)d_in[4];
    const float* W2 = (const float*)d_in[5];
    const float* b2 = (const float*)d_in[6];
    float*       out = (float*)d_out;

    const int nNodes = in_sizes[0] / F_IN;   // 100000
    const int nEdges = in_sizes[2];          // 3200000
    const int* srcIdx = ei;
    const int* dstIdx = ei + nEdges;

    float* ws      = (float*)d_ws;
    float* h_pre   = ws;                                   // N*128
    float* agg1    = h_pre   + (size_t)nNodes * HID;       // N*128
    float* out_pre = agg1    + (size_t)nNodes * HID;       // N*40
    float* agg2    = out_pre + (size_t)nNodes * NCLS;      // N*40

    const long long n1 = (long long)nNodes * HID;
    const long long n2 = (long long)nNodes * NCLS;

    zero_f32<<<2048, 256, 0, stream>>>(agg1, n1);
    zero_f32<<<2048, 256, 0, stream>>>(agg2, n2);

    // 32 rows per block (N = 100000 = 3125 * 32)
    gemm1_wmma<<<nNodes / 32, 256, 0, stream>>>(x, W1, h_pre);

    // one wave per edge
    {
        long long warps = nEdges;
        int blocks = (int)((warps * 32 + 255) / 256);
        scatter1<<<blocks, 256, 0, stream>>>(h_pre, srcIdx, dstIdx, ew, agg1, nEdges);
    }

    gemm2_wmma<<<nNodes / 32, 192, 0, stream>>>(agg1, b1, W2, out_pre);

    {
        long long total = (long long)nEdges * NCLS;
        int blocks = (int)((total + 255) / 256);
        scatter2<<<blocks, 256, 0, stream>>>(out_pre, srcIdx, dstIdx, ew, agg2, total);
    }

    logsoftmax<<<(nNodes + 255) / 256, 256, 0, stream>>>(agg2, b2, out, nNodes);
}